// MotionLayer_42314017800871
// MI455X (gfx1250) — compile-verified
//
#include <hip/hip_runtime.h>

// ---------------------------------------------------------------------------
// MotionLayer scan on MI455X (gfx1250).
//
// Roofline: 101 MB in + 201 MB out ~= 302 MB @ 23.3 TB/s -> ~13us floor.
// Strictly memory-bound, so the kernel is one-pass (read a once, write p,v
// once) and the scan math is moved onto the FP32 matrix core:
//   local cumsum        = L16 (lower-tri ones) @ X    (4 chained K=4 WMMAs)
//   local double-cumsum = M16 (M[t,s]=t-s+1)   @ X
// via V_WMMA_F32_16X16X4_F32 (exact fp32). Cross-chunk carries (S,D per
// series) are two scalars folded in with a wave32 lane shuffle.
//
// Data movement (CDNA5 async path): one wave owns one batch, so each
// 16-timestep input chunk is a contiguous 768B block. It is staged into LDS
// with double-buffered GLOBAL_LOAD_ASYNC_TO_LDS_B64 (ASYNCcnt +
// s_wait_asynccnt), prefetching chunk c+1 while chunk c computes, then read
// back bank-conflict-free with ds_load_b32 into WMMA B-operand layout.
// ---------------------------------------------------------------------------

typedef __attribute__((ext_vector_type(2))) float v2f;
typedef __attribute__((ext_vector_type(8))) float v8f;

#define NVEH   6
#define NC     (NVEH * 2)        // 12 scalar series per batch
#define TSTEPS 512
#define NBATCH 4096
#define DT_F   0.1f
#define WPB    8                 // wave32 per block
#define CHUNK_FLOATS (16 * NC)   // 192 floats = 768 bytes per time-chunk

static __device__ __forceinline__ v8f wmma_f32_16x16x4(v2f a, v2f b, v8f c) {
    // (neg_a, A, neg_b, B, c_mod, C, reuse_a, reuse_b)
    return __builtin_amdgcn_wmma_f32_16x16x4_f32(
        false, a, false, b, (short)0, c, false, false);
}

// Raw 32-bit LDS byte offset of a __shared__ object (for async-load VDST).
static __device__ __forceinline__ unsigned lds_offset(const float* p) {
    typedef __attribute__((address_space(3))) const float* lds_cfp;
    return (unsigned)(size_t)(lds_cfp)p;
}

// Stage one contiguous 768B chunk (global -> LDS) as 96 b64 segments,
// 3 per lane, tracked on ASYNCcnt.
static __device__ __forceinline__ void async_stage(const float* gsrc,
                                                   unsigned lds_byte_base,
                                                   int lane) {
    unsigned l = lds_byte_base + (unsigned)lane * 8u;
    const char* g = (const char*)gsrc + (size_t)lane * 8u;
#pragma unroll
    for (int i = 0; i < 3; ++i) {
        asm volatile("global_load_async_to_lds_b64 %0, %1, off"
                     :: "v"(l), "v"((const void*)g)
                     : "memory");
        l += 256u;
        g += 256;
    }
}

__global__ __launch_bounds__(256) void motion_scan_wmma(
    const float* __restrict__ accel,   // [B, T, 12]
    const float* __restrict__ init,    // [B, 4] (p0x,p0y,v0x,v0y of vehicle 0)
    float* __restrict__ out)           // [B, T, 24]
{
    __shared__ float smem[WPB * 2 * CHUNK_FLOATS];   // 12 KB / block

    const int lane = threadIdx.x & 31;
    const int widb = threadIdx.x >> 5;                       // wave in block
    const int b    = (int)blockIdx.x * WPB + widb;           // batch id
    const int col  = lane & 15;   // WMMA column: 0..11 real series, 12..15 dead
    const int hi   = lane >> 4;   // half-wave select
    const int ncol = (col < NC) ? col : (NC - 1);            // clamped series

    // --- constant triangular A operands (16x16 as four 16x4 K-chunks) ------
    // f32 A layout: lane holds row M = lane%16; VGPR j holds K = j + 2*hi.
    v2f AL[4], AM[4];
#pragma unroll
    for (int k = 0; k < 4; ++k) {
#pragma unroll
        for (int j = 0; j < 2; ++j) {
            const int kk = 4 * k + j + 2 * hi;
            AL[k][j] = (kk <= col) ? 1.0f : 0.0f;
            AM[k][j] = (kk <= col) ? (float)(col - kk + 1) : 0.0f;
        }
    }

    // --- per-lane series state --------------------------------------------
    const int veh = ncol >> 1;
    const int crd = ncol & 1;
    const float p0 = (ncol < 2) ? init[b * 4 + ncol]     : 0.0f;
    const float v0 = (ncol < 2) ? init[b * 4 + 2 + ncol] : 0.0f;

    const float dt   = DT_F;
    const float dt2  = DT_F * DT_F;
    const float hdt2 = 0.5f * DT_F * DT_F;
    const float v0dt = v0 * dt;
    const float hif  = 8.0f * (float)hi;   // timestep offset of the hi half

    const float* __restrict__ gbase = accel + (size_t)b * TSTEPS * NC;
    float* __restrict__ optr = out + (size_t)b * TSTEPS * 24 + veh * 4 + crd;

    // double-buffer in LDS
    float* buf0 = &smem[widb * 2 * CHUNK_FLOATS];
    float* buf1 = buf0 + CHUNK_FLOATS;
    const unsigned o0 = lds_offset(buf0);
    const unsigned o1 = lds_offset(buf1);

    float Sprev = 0.0f;   // running sum of a
    float Dprev = 0.0f;   // running sum of S
    float c0f   = 0.0f;   // float copy of chunk base timestep

    const int nchunks = TSTEPS / 16;   // 32

    // prime the pipeline: stage chunk 0
    async_stage(gbase, o0, lane);

    for (int c = 0; c < nchunks; ++c) {
        const float* buf = (c & 1) ? buf1 : buf0;

        if (c + 1 < nchunks) {
            // prefetch next chunk into the other buffer, then wait for the
            // 3 outstanding copies of THIS chunk (<=3 remain in flight).
            async_stage(gbase + (size_t)(c + 1) * CHUNK_FLOATS,
                        (c & 1) ? o0 : o1, lane);
            asm volatile("s_wait_asynccnt 0x3" ::: "memory");
        } else {
            asm volatile("s_wait_asynccnt 0x0" ::: "memory");
        }

        // ---- gather B operands from LDS (conflict-free ds_load_b32) ------
        // B layout per K=4 chunk: VGPR j holds row t = 4k+j+2*hi, col = lane%16
        v2f Bm[4];
#pragma unroll
        for (int k = 0; k < 4; ++k) {
#pragma unroll
            for (int j = 0; j < 2; ++j) {
                const int tl = 4 * k + j + 2 * hi;
                Bm[k][j] = buf[tl * NC + ncol];
            }
        }

        // ---- chained-WMMA blocked scan over K=16 -------------------------
        v8f s = {0.f, 0.f, 0.f, 0.f, 0.f, 0.f, 0.f, 0.f};
        v8f d = {0.f, 0.f, 0.f, 0.f, 0.f, 0.f, 0.f, 0.f};
#pragma unroll
        for (int k = 0; k < 4; ++k) {
            s = wmma_f32_16x16x4(AL[k], Bm[k], s);   // local cumsum
            d = wmma_f32_16x16x4(AM[k], Bm[k], d);   // local double cumsum
        }

        // ---- fold carries, emit p and v (only the 12 real columns) -------
        // C/D layout: VGPR r -> local timestep r + 8*hi, column = lane%16
        if (col < NC) {
            float* oc = optr + (size_t)((int)c0f + 8 * hi) * 24;
#pragma unroll
            for (int r = 0; r < 8; ++r) {
                const float tl1 = (float)(r + 1) + hif;   // steps into tile
                const float tau = c0f + tl1;              // 1-indexed step
                const float Sg  = Sprev + s[r];
                const float Dg  = Dprev + tl1 * Sprev + d[r];
                oc[0] = p0 + v0dt * tau + dt2 * Dg + hdt2 * Sg;  // position
                oc[2] = v0 + dt * Sg;                            // velocity
                oc += 24;
            }
        }

        // ---- advance carries: tile totals sit at local t=15 (VGPR7, hi) --
        const float sTot = __shfl(s[7], col + 16, 32);
        const float dTot = __shfl(d[7], col + 16, 32);
        Dprev += 16.0f * Sprev + dTot;
        Sprev += sTot;
        c0f   += 16.0f;
    }
}

extern "C" void kernel_launch(void* const* d_in, const int* in_sizes, int n_in,
                              void* d_out, int out_size, void* d_ws, size_t ws_size,
                              hipStream_t stream) {
    (void)in_sizes; (void)n_in; (void)out_size; (void)d_ws; (void)ws_size;
    const float* accel = (const float*)d_in[0];   // [4096, 512, 12] f32
    const float* init  = (const float*)d_in[1];   // [4096, 4] f32
    float* out = (float*)d_out;                   // [4096, 512, 24] f32

    const int threads = 32 * WPB;            // 256 = 8 wave32
    const int blocks  = NBATCH / WPB;        // 512 blocks, 1 wave per batch

    motion_scan_wmma<<<blocks, threads, 0, stream>>>(accel, init, out);
}